// CRF_77936476554057
// MI455X (gfx1250) — compile-verified
//
#include <hip/hip_runtime.h>
#include <hip/hip_bf16.h>

// CRF forward (log-partition) for MI455X / gfx1250.
//
// (LSE,+) semiring scan. Phase 1: 32 batches x 16 chunks of 32 steps; each
// workgroup folds its chunk's 64x64 transition matrices into one semiring
// product, kept in scaled-linear space so each step is a real f32 64x64x64
// GEMM on V_WMMA_F32_16X16X4_F32 (16 waves, one 16x16 C tile each, 16 K-slices
// of depth 4). E_i tiles (16 KB) are double-buffered in LDS via the Tensor
// Data Mover (tensor_load_to_lds, TENSORcnt), so step i+1's HBM traffic hides
// behind step i's exp+WMMA work. Per-step scale fixups cost 64 exps; per-row
// renorm every 8 steps costs 64 logs. Phase 2 folds 16 chunk matrices into
// alpha0 per batch + gold gather. Phase 3: deterministic scalar reduction.

#define L 64
#define S_LEN 512
#define B_SZ 32
#define NCHUNK 16
#define CSZ 32      // steps per chunk (covers i = 1 .. 511)

typedef __attribute__((ext_vector_type(2))) float        v2f;
typedef __attribute__((ext_vector_type(8))) float        v8f;
typedef __attribute__((ext_vector_type(4))) unsigned int u32x4;
typedef __attribute__((ext_vector_type(4))) int          i32x4;
typedef __attribute__((ext_vector_type(8))) int          i32x8;

#if __has_builtin(__builtin_amdgcn_tensor_load_to_lds) && \
    __has_builtin(__builtin_amdgcn_s_wait_tensorcnt)
#define HAVE_TDM 1
#else
#define HAVE_TDM 0
#endif

#if HAVE_TDM
// One-descriptor TDM transfer: 16 KB contiguous tile (4096 x f32, 1 row).
// D# packing per CDNA5 ISA §8.3/8.4. lds pointer: LDS aperture guarantees
// addr[31:0] == LDS byte offset, so the low 32 bits are D#.lds_addr.
__device__ __forceinline__ void tdm_load_tile(const float* gsrc, float* ldst)
{
    const unsigned long long ga = (unsigned long long)(uintptr_t)gsrc;
    u32x4 g0;
    g0[0] = 1u;                                    // count=1, user descriptor
    g0[1] = (unsigned int)(uintptr_t)ldst;         // lds_addr (bytes)
    g0[2] = (unsigned int)ga;                      // global_addr[31:0]
    g0[3] = (unsigned int)((ga >> 32) & 0x1FFFFFFu) | (2u << 30); // [56:32], type=2
    i32x8 g1;
    g1[0] = (int)(2u << 16);                       // data_size = 4 bytes
    g1[1] = (int)(4096u << 16);                    // tensor_dim0 = 4096 (lo16)
    g1[2] = (int)(1u << 16);                       // tensor_dim0 hi=0 | tensor_dim1=1
    g1[3] = (int)(4096u << 16);                    // tensor_dim1 hi=0 | tile_dim0=4096
    g1[4] = 1;                                     // tile_dim1=1, tile_dim2=0
    g1[5] = 4096;                                  // tensor_dim0_stride lo32
    g1[6] = (int)(4096u << 16);                    // tensor_dim1_stride lo16
    g1[7] = 0;
    i32x4 z4 = {0, 0, 0, 0};                       // groups 2/3: <=2D tensor
#if defined(__clang_major__) && (__clang_major__ >= 23)
    i32x8 z8 = {0, 0, 0, 0, 0, 0, 0, 0};
    __builtin_amdgcn_tensor_load_to_lds(g0, g1, z4, z4, z8, 0);
#else
    __builtin_amdgcn_tensor_load_to_lds(g0, g1, z4, z4, 0);
#endif
}
#endif

// ---------------------------------------------------------------- Phase 1 ---
__global__ __launch_bounds__(512)
void crf_chunk_kernel(const float* __restrict__ emits,
                      const unsigned char* __restrict__ mask,
                      float* __restrict__ chunkM)   // [B][NCHUNK][64][64]
{
    __shared__ float P[L * L];       // running scaled-linear product
    __shared__ float E0[L * L];      // double-buffered E_i tiles
    __shared__ float E1[L * L];
    __shared__ float cmax[L];
    __shared__ float fcol[L];
    __shared__ float mrow[L];
    __shared__ float rmax[L];
    __shared__ float msum;
    __shared__ int   len;

    const int b = blockIdx.x >> 4;      // / NCHUNK
    const int c = blockIdx.x & (NCHUNK - 1);
    const int t = threadIdx.x;          // 0..511

    if (t == 0) { len = 0; msum = 0.0f; }
    __syncthreads();
    // mask is prefix-true per row, so popcount == length
    if (mask[b * S_LEN + t]) atomicAdd(&len, 1);
    __syncthreads();

    const int i0   = 1 + c * CSZ;
    const int i1   = min(1 + (c + 1) * CSZ, S_LEN);
    const int iend = min(i1, len);      // steps apply only while i < len_b

    const int wave = t >> 5, lane = t & 31;

#if HAVE_TDM
    // Prologue: kick off the first tile ASAP (one DMA per workgroup; TDM
    // ignores EXEC and issues once per wave -> gate on wave 0 only).
    if (wave == 0 && i0 < iend)
        tdm_load_tile(emits + (((size_t)b * S_LEN + i0) << 12), E0);
#endif

    // P = identity (semiring identity after final log: diag 0, off-diag -inf)
#pragma unroll
    for (int e = 0; e < 8; ++e) {
        int idx = t * 8 + e;
        P[idx] = ((idx >> 6) == (idx & 63)) ? 1.0f : 0.0f;
    }
    if (t < L) mrow[t] = 0.0f;
    __syncthreads();

    const int ta = wave >> 2, tn = wave & 3;       // 4x4 tile grid of C
    const int rsub = lane & 15, kh = lane >> 4;    // f32 A/B fragment layout

    for (int i = i0; i < iend; ++i) {
        float* Ecur = (((i - i0) & 1) == 0) ? E0 : E1;
        float* Enxt = (((i - i0) & 1) == 0) ? E1 : E0;

#if HAVE_TDM
        // Wait for E_i to land (issuing wave), publish to all waves, then
        // immediately start the DMA for E_{i+1} into the other buffer.
        if (wave == 0)
            __builtin_amdgcn_s_wait_tensorcnt(0);
        __syncthreads();
        if (wave == 0 && (i + 1) < iend)
            tdm_load_tile(emits + (((size_t)b * S_LEN + i + 1) << 12), Enxt);
#else
        // Fallback: synchronous cooperative staging + speculative prefetch.
        {
            const float4* src = (const float4*)(emits + (((size_t)b * S_LEN + i) << 12));
            float4* Ev = (float4*)Ecur;
            Ev[t]       = src[t];
            Ev[t + 512] = src[t + 512];
            if (i + 1 < iend) {
                const char* nxt = (const char*)(emits + (((size_t)b * S_LEN + i + 1) << 12));
                __builtin_prefetch(nxt + ((size_t)t << 5), 0, 0);
            }
            (void)Enxt;
        }
        __syncthreads();
#endif

        // ---- per-column max (bank-conflict-free: thread j owns bank j) ----
        if (t < L) {
            float m = Ecur[t];
            for (int k = 1; k < L; ++k) m = fmaxf(m, Ecur[k * L + t]);
            cmax[t] = m;
        }
        __syncthreads();

        // ---- fcol[j] = exp(cmax_j - s), msum += s (only 64 exps/step) -----
        if (t < L) {
            float s = cmax[0];
            for (int j = 1; j < L; ++j) s = fmaxf(s, cmax[j]);
            fcol[t] = __expf(cmax[t] - s);
            if (t == 0) msum += s;
        }
        // ---- E := exp(E - cmax_j)  (the one-exp-per-input-element cost) ---
#pragma unroll
        for (int e = 0; e < 8; ++e) {
            int idx = t * 8 + e;
            Ecur[idx] = __expf(Ecur[idx] - cmax[idx & 63]);
        }
        __syncthreads();

        // ---- Q = P x E via v_wmma_f32_16x16x4_f32, one 16x16 tile/wave ----
        v8f acc = {0.f, 0.f, 0.f, 0.f, 0.f, 0.f, 0.f, 0.f};
#pragma unroll
        for (int kk = 0; kk < 16; ++kk) {
            const int kbase = kk * 4 + kh * 2;
            v2f a, bb;
            a.x  = P[(ta * 16 + rsub) * L + kbase];
            a.y  = P[(ta * 16 + rsub) * L + kbase + 1];
            bb.x = Ecur[(kbase)     * L + tn * 16 + rsub];
            bb.y = Ecur[(kbase + 1) * L + tn * 16 + rsub];
            acc = __builtin_amdgcn_wmma_f32_16x16x4_f32(
                false, a, false, bb, (short)0, acc, false, false);
        }
        __syncthreads();   // all waves done reading P & Ecur

        // ---- P := Q * fcol[j]  (C/D layout: lanes 16..31 hold rows M+8) ---
        {
            const int col = tn * 16 + (lane & 15);
            const int mo  = (lane >> 4) * 8;
            const float f = fcol[col];
#pragma unroll
            for (int r = 0; r < 8; ++r)
                P[(ta * 16 + mo + r) * L + col] = acc[r] * f;
        }
        __syncthreads();

        // ---- row renorm every 8 steps: only 64 maxes + 64 logs -----------
        if ((((i - i0) & 7) == 7) || (i == iend - 1)) {
            if (t < L) {
                float m = 0.0f;
                for (int j = 0; j < L; ++j) m = fmaxf(m, P[t * L + j]);
                rmax[t] = m;
                mrow[t] += __logf(m);
            }
            __syncthreads();
#pragma unroll
            for (int e = 0; e < 8; ++e) {
                int idx = t * 8 + e;
                P[idx] *= 1.0f / rmax[idx >> 6];
            }
            __syncthreads();
        }
    }

    __syncthreads();
    // logM[a,j] = msum + mrow[a] + log P[a,j]   (log 0 -> -inf: identity ok)
    const float ms = msum;
    float* out = chunkM + ((size_t)(b * NCHUNK + c) << 12);
#pragma unroll
    for (int e = 0; e < 8; ++e) {
        int idx = t * 8 + e;
        out[idx] = ms + mrow[idx >> 6] + __logf(P[idx]);
    }
}

// ---------------------------------------------------------------- Phase 2 ---
__global__ __launch_bounds__(64)
void crf_combine_kernel(const float* __restrict__ emits,
                        const int* __restrict__ targets,
                        const unsigned char* __restrict__ mask,
                        const float* __restrict__ chunkM,
                        float* __restrict__ partials)   // [B][3]
{
    __shared__ float alpha[L];
    __shared__ float red[L];
    const int b = blockIdx.x;
    const int j = threadIdx.x;   // 0..63

    // alpha0 = E_0[BOS=0, :]
    alpha[j] = emits[(((size_t)b * S_LEN + 0) << 12) + j];
    __syncthreads();

    for (int c = 0; c < NCHUNK; ++c) {
        const float* M = chunkM + ((size_t)(b * NCHUNK + c) << 12);
        float m = -INFINITY;
        for (int k = 0; k < L; ++k) m = fmaxf(m, alpha[k] + M[k * L + j]);
        float na;
        if (m == -INFINITY) {
            na = -INFINITY;
        } else {
            float ss = 0.0f;
            for (int k = 0; k < L; ++k) ss += __expf(alpha[k] + M[k * L + j] - m);
            na = m + __logf(ss);
        }
        __syncthreads();
        alpha[j] = na;
        __syncthreads();
    }

    // gold score + token count for this batch
    float gold = 0.0f;
    float cnt  = 0.0f;
    for (int i = j; i < S_LEN; i += L) {
        if (mask[b * S_LEN + i]) {
            int t0 = targets[b * (S_LEN + 1) + i];
            int t1 = targets[b * (S_LEN + 1) + i + 1];
            gold += emits[(((size_t)b * S_LEN + i) << 12) + t0 * L + t1];
            cnt  += 1.0f;
        }
    }

    // deterministic fixed-order reductions by thread 0
    if (j == 0) {
        float m = alpha[0];
        for (int k = 1; k < L; ++k) m = fmaxf(m, alpha[k]);
        float ss = 0.0f;
        for (int k = 0; k < L; ++k) ss += __expf(alpha[k] - m);
        partials[b * 3 + 0] = m + __logf(ss);      // log_z_b
    }
    red[j] = gold;
    __syncthreads();
    if (j == 0) {
        float g = 0.0f;
        for (int k = 0; k < L; ++k) g += red[k];
        partials[b * 3 + 1] = g;
    }
    __syncthreads();
    red[j] = cnt;
    __syncthreads();
    if (j == 0) {
        float n = 0.0f;
        for (int k = 0; k < L; ++k) n += red[k];
        partials[b * 3 + 2] = n;
    }
}

// ---------------------------------------------------------------- Phase 3 ---
__global__ void crf_finalize_kernel(const float* __restrict__ partials,
                                    float* __restrict__ out)
{
    float lz = 0.0f, gold = 0.0f, cnt = 0.0f;
    for (int b = 0; b < B_SZ; ++b) {
        lz   += partials[b * 3 + 0];
        gold += partials[b * 3 + 1];
        cnt  += partials[b * 3 + 2];
    }
    out[0] = (lz - gold) / cnt;
}

// ----------------------------------------------------------------- launch ---
extern "C" void kernel_launch(void* const* d_in, const int* in_sizes, int n_in,
                              void* d_out, int out_size, void* d_ws, size_t ws_size,
                              hipStream_t stream) {
    const float*         emits   = (const float*)d_in[0];
    const int*           targets = (const int*)d_in[1];          // int per harness
    const unsigned char* mask    = (const unsigned char*)d_in[2]; // numpy bool = 1B

    float* chunkM   = (float*)d_ws;                               // 8 MB
    float* partials = chunkM + (size_t)B_SZ * NCHUNK * L * L;     // 96 floats

    crf_chunk_kernel<<<B_SZ * NCHUNK, 512, 0, stream>>>(emits, mask, chunkM);
    crf_combine_kernel<<<B_SZ, 64, 0, stream>>>(emits, targets, mask, chunkM, partials);
    crf_finalize_kernel<<<1, 1, 0, stream>>>(partials, (float*)d_out);
}